// HybridKernelLSTM_65481071406125
// MI455X (gfx1250) — compile-verified
//
#include <hip/hip_runtime.h>
#include <hip/hip_bf16.h>
#include <math.h>

typedef float v2f __attribute__((ext_vector_type(2)));
typedef float v8f __attribute__((ext_vector_type(8)));

#define WMMA4(a, b, c) \
    __builtin_amdgcn_wmma_f32_16x16x4_f32(false, (a), false, (b), (short)0, (c), false, false)

// Problem constants
#define TB   8192      // T*B = 128*64 rows of xf
#define NS   1024      // support vectors
#define DD   32        // feature dim
#define NT   64        // tiles along N (1024/16)
#define MT   512       // tiles along M (8192/16)

__device__ __forceinline__ float sigmoidf_(float x) { return 1.0f / (1.0f + expf(-x)); }

// ---------------------------------------------------------------------------
// Kernel 1: per-row prep.
//   For each xf row (8192) and each support vector (1024):
//     - squared norm (for RBF)
//     - 16-dim complex circuit state (product-state closed form + LINE perms)
// Product state: wire w gets RX(row[w]+qkp[w]); amp[i] = prod(c_w or s_w) * (-i)^popcount
// LINE perm composition: new[j] = orig[p01[p12[p23[j]]]]
// ---------------------------------------------------------------------------
__global__ void prep_kernel(const float* __restrict__ x, const float* __restrict__ sv,
                            const float* __restrict__ qkp,
                            float* __restrict__ sxr, float* __restrict__ sxi,
                            float* __restrict__ syr, float* __restrict__ syi,
                            float* __restrict__ xx, float* __restrict__ ss) {
    int tid = blockIdx.x * blockDim.x + threadIdx.x;
    if (tid >= TB + NS) return;

    const float* row;
    float *dr, *di, *nrm;
    if (tid < TB) {
        row = x + (size_t)tid * DD;
        dr = sxr + (size_t)tid * 16; di = sxi + (size_t)tid * 16; nrm = xx + tid;
    } else {
        int idx = tid - TB;
        row = sv + (size_t)idx * DD;
        dr = syr + (size_t)idx * 16; di = syi + (size_t)idx * 16; nrm = ss + idx;
    }

    float s2 = 0.f;
    #pragma unroll
    for (int k = 0; k < DD; ++k) s2 += row[k] * row[k];
    *nrm = s2;

    float cw[4], sw[4];
    #pragma unroll
    for (int w = 0; w < 4; ++w) {
        float th = 0.5f * (row[w] + qkp[w]);
        cw[w] = cosf(th);
        sw[w] = sinf(th);
    }

    float ar[16], ai[16];
    #pragma unroll
    for (int i = 0; i < 16; ++i) {
        float mag = 1.f;
        int pc = 0;
        #pragma unroll
        for (int w = 0; w < 4; ++w) {
            if (i & (1 << (3 - w))) { mag *= sw[w]; ++pc; }
            else                    { mag *= cw[w]; }
        }
        // (-i)^pc
        float re, im;
        switch (pc & 3) {
            case 0: re =  mag; im =  0.f; break;
            case 1: re =  0.f; im = -mag; break;
            case 2: re = -mag; im =  0.f; break;
            default: re = 0.f; im =  mag; break;
        }
        ar[i] = re; ai[i] = im;
    }

    // LINE perms: chain j -> p23 -> p12 -> p01
    #pragma unroll
    for (int j = 0; j < 16; ++j) {
        int q = j;
        q = (q & 2) ? (q ^ 1) : q;  // cnot(2,3)
        q = (q & 4) ? (q ^ 2) : q;  // cnot(1,2)
        q = (q & 8) ? (q ^ 4) : q;  // cnot(0,1)
        dr[j] = ar[q];
        di[j] = ai[q];
    }
}

// ---------------------------------------------------------------------------
// Kernel 2: RBF via WMMA f32 16x16x4. One wave -> one 16x16 tile.
//   dot = X(16x32) . SV^T(32x16), rbf = exp(-(|x|^2 + |s|^2 - 2 dot))
// Fragment addressing (A and B symmetric): lane L, comp j holds
//   K = 4*kc + 2*(L/16) + j at row/col (L%16)  => one float2 load each.
// ---------------------------------------------------------------------------
__global__ void rbf_kernel(const float* __restrict__ X, const float* __restrict__ SV,
                           const float* __restrict__ xx, const float* __restrict__ ss,
                           float* __restrict__ out) {
    int wave  = blockIdx.x * (blockDim.x >> 5) + (threadIdx.x >> 5);
    int tileN = wave & (NT - 1);
    int tileM = wave >> 6;          // / NT
    int lane  = threadIdx.x & 31;
    int half  = lane >> 4;
    int mr    = lane & 15;

    const float* xrow = X  + (size_t)(tileM * 16 + mr) * DD;
    const float* srow = SV + (size_t)(tileN * 16 + mr) * DD;

    v8f acc = {0.f, 0.f, 0.f, 0.f, 0.f, 0.f, 0.f, 0.f};
    #pragma unroll
    for (int kc = 0; kc < 8; ++kc) {
        v2f a = *(const v2f*)(xrow + kc * 4 + 2 * half);
        v2f b = *(const v2f*)(srow + kc * 4 + 2 * half);
        acc = WMMA4(a, b, acc);
    }

    float ssn = ss[tileN * 16 + mr];
    int col = tileN * 16 + mr;
    #pragma unroll
    for (int r = 0; r < 8; ++r) {
        int row = tileM * 16 + r + 8 * half;
        float d2 = xx[row] + ssn - 2.0f * acc[r];
        out[(size_t)row * NS + col] = expf(-d2);
    }
}

// ---------------------------------------------------------------------------
// Kernel 3: qk = |sx . sy^T| via 4 real WMMA GEMMs (K=16).
//   Re = Ar.Br^T - Ai.Bi^T ; Im = Ar.Bi^T + Ai.Br^T
// f32 WMMA has no A-neg modifier -> negate Ai fragment in VALU.
// ---------------------------------------------------------------------------
__global__ void qk_kernel(const float* __restrict__ sxr, const float* __restrict__ sxi,
                          const float* __restrict__ syr, const float* __restrict__ syi,
                          float* __restrict__ out) {
    int wave  = blockIdx.x * (blockDim.x >> 5) + (threadIdx.x >> 5);
    int tileN = wave & (NT - 1);
    int tileM = wave >> 6;
    int lane  = threadIdx.x & 31;
    int half  = lane >> 4;
    int mr    = lane & 15;

    const float* arow = sxr + (size_t)(tileM * 16 + mr) * 16;
    const float* irow = sxi + (size_t)(tileM * 16 + mr) * 16;
    const float* brow = syr + (size_t)(tileN * 16 + mr) * 16;
    const float* jrow = syi + (size_t)(tileN * 16 + mr) * 16;

    v8f accR = {0.f, 0.f, 0.f, 0.f, 0.f, 0.f, 0.f, 0.f};
    v8f accI = {0.f, 0.f, 0.f, 0.f, 0.f, 0.f, 0.f, 0.f};
    #pragma unroll
    for (int kc = 0; kc < 4; ++kc) {
        int off = kc * 4 + 2 * half;
        v2f axr = *(const v2f*)(arow + off);
        v2f axi = *(const v2f*)(irow + off);
        v2f byr = *(const v2f*)(brow + off);
        v2f byi = *(const v2f*)(jrow + off);
        v2f naxi = -axi;
        accR = WMMA4(axr,  byr, accR);
        accR = WMMA4(naxi, byi, accR);   // - Ai.Bi
        accI = WMMA4(axr,  byi, accI);
        accI = WMMA4(axi,  byr, accI);
    }

    int col = tileN * 16 + mr;
    #pragma unroll
    for (int r = 0; r < 8; ++r) {
        int row = tileM * 16 + r + 8 * half;
        float re = accR[r], im = accI[r];
        out[(size_t)row * NS + col] = sqrtf(re * re + im * im);
    }
}

// ---------------------------------------------------------------------------
// Kernel 4: persistent-block LSTM scan (T=128, B=64, H=4).
// Thread t: batch b = t>>2, gate g = t&3 (order f,i,u,o).
// Gate = measure_z(circuit(comb.W^T + b, p, WRAP)): product-state probabilities
// (phases irrelevant for Z-measure), WRAP perm chain j -> p30 -> p23 -> p12 -> p01.
// g==0 threads own the c state and do the elementwise combine.
// ---------------------------------------------------------------------------
__global__ void lstm_kernel(const float* __restrict__ x,
                            const float* __restrict__ Wf, const float* __restrict__ bf,
                            const float* __restrict__ Wi, const float* __restrict__ bi,
                            const float* __restrict__ Wu, const float* __restrict__ bu,
                            const float* __restrict__ Wo, const float* __restrict__ bo,
                            const float* __restrict__ pf, const float* __restrict__ pig,
                            const float* __restrict__ pu, const float* __restrict__ po,
                            float* __restrict__ outs) {
    __shared__ float Wl[4][4][36];   // [gate][out][in]
    __shared__ float bl[4][4];
    __shared__ float pl[4][4];
    __shared__ float h_lds[64][4];
    __shared__ float gv[64][4][4];   // [batch][gate][comp]

    int t = threadIdx.x;             // 0..255
    for (int idx = t; idx < 4 * 144; idx += 256) {
        int g = idx / 144, r = idx % 144;
        const float* W = (g == 0) ? Wf : (g == 1) ? Wi : (g == 2) ? Wu : Wo;
        Wl[g][r / 36][r % 36] = W[r];
    }
    if (t < 16) {
        int g = t / 4, k = t % 4;
        const float* bb = (g == 0) ? bf : (g == 1) ? bi : (g == 2) ? bu : bo;
        const float* pp = (g == 0) ? pf : (g == 1) ? pig : (g == 2) ? pu : po;
        bl[g][k] = bb[k];
        pl[g][k] = pp[k];
    }
    ((float*)h_lds)[t] = 0.f;
    __syncthreads();

    int b = t >> 2, g = t & 3;
    float c0 = 0.f, c1 = 0.f, c2 = 0.f, c3 = 0.f;   // cell state (g==0 threads)

    for (int step = 0; step < 128; ++step) {
        const float* xt = x + (size_t)(step * 64 + b) * 32;

        float ang[4];
        #pragma unroll
        for (int j = 0; j < 4; ++j) {
            float a = bl[g][j];
            for (int k = 0; k < 32; ++k) a += xt[k] * Wl[g][j][k];
            #pragma unroll
            for (int k = 0; k < 4; ++k) a += h_lds[b][k] * Wl[g][j][32 + k];
            ang[j] = a;
        }

        float cw[4], sw[4];
        #pragma unroll
        for (int w = 0; w < 4; ++w) {
            float th = 0.5f * (ang[w] + pl[g][w]);
            float cc = cosf(th), ssn = sinf(th);
            cw[w] = cc * cc;
            sw[w] = ssn * ssn;
        }

        float m0 = 0.f, m1 = 0.f, m2 = 0.f, m3 = 0.f;
        #pragma unroll
        for (int j = 0; j < 16; ++j) {
            int q = j;
            q = (q & 1) ? (q ^ 8) : q;  // cnot(3,0) (applied last -> first in chain)
            q = (q & 2) ? (q ^ 1) : q;  // cnot(2,3)
            q = (q & 4) ? (q ^ 2) : q;  // cnot(1,2)
            q = (q & 8) ? (q ^ 4) : q;  // cnot(0,1)
            float p = 1.f;
            #pragma unroll
            for (int w = 0; w < 4; ++w) p *= (q & (1 << (3 - w))) ? sw[w] : cw[w];
            m0 += (j & 8) ? -p : p;
            m1 += (j & 4) ? -p : p;
            m2 += (j & 2) ? -p : p;
            m3 += (j & 1) ? -p : p;
        }
        gv[b][g][0] = m0; gv[b][g][1] = m1; gv[b][g][2] = m2; gv[b][g][3] = m3;
        __syncthreads();

        if (g == 0) {
            float cc[4] = {c0, c1, c2, c3};
            #pragma unroll
            for (int k = 0; k < 4; ++k) {
                float fv = sigmoidf_(gv[b][0][k]);
                float iv = sigmoidf_(gv[b][1][k]);
                float uv = tanhf(gv[b][2][k]);
                float ov = sigmoidf_(gv[b][3][k]);
                float cn = fv * cc[k] + iv * uv;
                cc[k] = cn;
                float hv = ov * tanhf(cn);
                h_lds[b][k] = hv;
                outs[(size_t)(step * 64 + b) * 4 + k] = hv;
            }
            c0 = cc[0]; c1 = cc[1]; c2 = cc[2]; c3 = cc[3];
        }
        __syncthreads();
    }
}

// ---------------------------------------------------------------------------
extern "C" void kernel_launch(void* const* d_in, const int* in_sizes, int n_in,
                              void* d_out, int out_size, void* d_ws, size_t ws_size,
                              hipStream_t stream) {
    const float* x   = (const float*)d_in[0];   // (128,64,32)
    const float* sv  = (const float*)d_in[1];   // (1024,32)
    const float* Wf  = (const float*)d_in[2];
    const float* bf  = (const float*)d_in[3];
    const float* Wi  = (const float*)d_in[4];
    const float* bi  = (const float*)d_in[5];
    const float* Wu  = (const float*)d_in[6];
    const float* bu  = (const float*)d_in[7];
    const float* Wo  = (const float*)d_in[8];
    const float* bo  = (const float*)d_in[9];
    const float* pf  = (const float*)d_in[10];
    const float* pig = (const float*)d_in[11];
    const float* pu  = (const float*)d_in[12];
    const float* po  = (const float*)d_in[13];
    const float* qkp = (const float*)d_in[14];

    float* out  = (float*)d_out;
    float* outs = out;                              // 128*64*4 = 32768
    float* rbf  = out + 32768;                      // 8192*1024
    float* qk   = out + 32768 + (size_t)TB * NS;    // 8192*1024

    float* ws   = (float*)d_ws;
    float* sxr = ws;                        // 8192*16
    float* sxi = sxr + (size_t)TB * 16;     // 8192*16
    float* syr = sxi + (size_t)TB * 16;     // 1024*16
    float* syi = syr + (size_t)NS * 16;     // 1024*16
    float* xx  = syi + (size_t)NS * 16;     // 8192
    float* ss  = xx + TB;                   // 1024

    // 1) per-row states + norms
    prep_kernel<<<(TB + NS + 255) / 256, 256, 0, stream>>>(x, sv, qkp,
                                                           sxr, sxi, syr, syi, xx, ss);

    // 2) RBF GEMM: 512x64 tiles, 8 waves/block -> 4096 blocks
    rbf_kernel<<<(MT * NT) / 8, 256, 0, stream>>>(x, sv, xx, ss, rbf);

    // 3) complex GEMM + abs
    qk_kernel<<<(MT * NT) / 8, 256, 0, stream>>>(sxr, sxi, syr, syi, qk);

    // 4) sequential LSTM scan, single persistent block
    lstm_kernel<<<1, 256, 0, stream>>>(x, Wf, bf, Wi, bi, Wu, bu, Wo, bo,
                                       pf, pig, pu, po, outs);
}